// Net_6425271075378
// MI455X (gfx1250) — compile-verified
//
#include <hip/hip_runtime.h>
#include <math.h>

// ---------------------------------------------------------------------------
// SplineCNN forward for gfx1250 (MI455X).  FP32 end-to-end; matrix work via
// V_WMMA_F32_16X16X4_F32 (wave32, 16x16 tiles, K stepped by 4).
// ---------------------------------------------------------------------------

typedef __attribute__((ext_vector_type(2))) float v2f;
typedef __attribute__((ext_vector_type(8))) float v8f;

__device__ __forceinline__ float eluf(float x) {
  return x > 0.f ? x : (expf(x) - 1.f);
}

// order-preserving float -> uint map for atomic max pooling
__device__ __forceinline__ unsigned f2ord(float f) {
  unsigned u = __float_as_uint(f);
  return (u & 0x80000000u) ? ~u : (u | 0x80000000u);
}
__device__ __forceinline__ float ord2f(unsigned u) {
  if (u == 0u) return 0.f;  // empty segment -> reference maps -inf to 0
  return (u & 0x80000000u) ? __uint_as_float(u ^ 0x80000000u)
                           : __uint_as_float(~u);
}

// --------------------------- elementwise helpers ---------------------------

__global__ void k_zero(float* p, int n) {
  int t = blockIdx.x * blockDim.x + threadIdx.x;
  if (t < n) p[t] = 0.f;
}

// Per edge: 8 trilinear corners (kernel index + basis weight), plus degree.
__global__ void k_edge_prep(const float* __restrict__ pseudo,
                            const int* __restrict__ dst, int E,
                            int* __restrict__ kidx8, float* __restrict__ basis8,
                            float* __restrict__ deg) {
  int e = blockIdx.x * blockDim.x + threadIdx.x;
  if (e >= E) return;
  float v0 = pseudo[e * 3 + 0] * 4.f;
  float v1 = pseudo[e * 3 + 1] * 4.f;
  float v2 = pseudo[e * 3 + 2] * 4.f;
  float l0 = fminf(fmaxf(floorf(v0), 0.f), 3.f);
  float l1 = fminf(fmaxf(floorf(v1), 0.f), 3.f);
  float l2 = fminf(fmaxf(floorf(v2), 0.f), 3.f);
  float f0 = v0 - l0, f1 = v1 - l1, f2 = v2 - l2;
  int i0 = (int)l0, i1 = (int)l1, i2 = (int)l2;
#pragma unroll
  for (int c = 0; c < 8; ++c) {
    int b0 = c & 1, b1 = (c >> 1) & 1, b2 = (c >> 2) & 1;
    int k = (i0 + b0) + (i1 + b1) * 5 + (i2 + b2) * 25;
    float w = (b0 ? f0 : 1.f - f0) * (b1 ? f1 : 1.f - f1) * (b2 ? f2 : 1.f - f2);
    kidx8[e * 8 + c] = k;
    basis8[e * 8 + c] = w;
  }
  atomicAdd(&deg[dst[e]], 1.0f);
}

// Layer 1 (in=1): msg[e,o] = x[src]*sum_c basis*W1[k,o]; atomic scatter to agg.
__global__ void k_l1_edge(const float* __restrict__ x0,
                          const int* __restrict__ src,
                          const int* __restrict__ dst,
                          const float* __restrict__ W1,
                          const int* __restrict__ kidx8,
                          const float* __restrict__ basis8,
                          float* __restrict__ agg, int E) {
  int t = blockIdx.x * blockDim.x + threadIdx.x;
  if (t >= E * 32) return;
  int o = t & 31;
  int e = t >> 5;
  float wsum = 0.f;
#pragma unroll
  for (int c = 0; c < 8; ++c)
    wsum += basis8[e * 8 + c] * W1[kidx8[e * 8 + c] * 32 + o];
  atomicAdd(&agg[dst[e] * 32 + o], x0[src[e]] * wsum);
}

// ------------------------- batched GEMM via WMMA ---------------------------
// Y[n, kk, o] = sum_i X[n,i] * Wk[kk][i,o] for kk in [0,kcount).
// One wave computes a 16x16 tile; K stepped by 4 (f32 WMMA).
__global__ void k_ygemm(const float* __restrict__ X, const float* __restrict__ W,
                        float* __restrict__ Y, int IN, int OUT, int kcount) {
  int ntiles = OUT >> 4;
  int bx = blockIdx.x;
  int nt = bx % ntiles; bx /= ntiles;
  int kk = bx % kcount; bx /= kcount;
  int mt = bx;
  const float* Wk = W + (size_t)kk * IN * OUT;
  int lane = threadIdx.x;
  int half = lane >> 4;   // 0: K pair {0,1}, 1: K pair {2,3}
  int lm = lane & 15;
  int arow = mt * 16 + lm;      // A-matrix row (m)
  int ncol = nt * 16 + lm;      // B/C column (n)
  v8f c = {};
  for (int k0 = 0; k0 < IN; k0 += 4) {
    int ka = k0 + 2 * half;
    v2f a = { X[(size_t)arow * IN + ka], X[(size_t)arow * IN + ka + 1] };
    v2f b = { Wk[(size_t)ka * OUT + ncol], Wk[(size_t)(ka + 1) * OUT + ncol] };
    c = __builtin_amdgcn_wmma_f32_16x16x4_f32(false, a, false, b,
                                              (short)0, c, false, false);
  }
#pragma unroll
  for (int v = 0; v < 8; ++v) {
    int m = mt * 16 + v + 8 * half;
    Y[(size_t)m * ((size_t)kcount * OUT) + (size_t)kk * OUT + ncol] = c[v];
  }
}

// Per (edge, out): accumulate basis-weighted gather from the current k-chunk.
__global__ void k_msg_accum(const float* __restrict__ Y,
                            const int* __restrict__ src,
                            const int* __restrict__ kidx8,
                            const float* __restrict__ basis8,
                            float* __restrict__ msg, int OUT, int kbase,
                            int kcount, int E, int first) {
  int t = blockIdx.x * blockDim.x + threadIdx.x;
  if (t >= E * OUT) return;
  int o = t % OUT;
  int e = t / OUT;
  const float* Yn = Y + (size_t)src[e] * ((size_t)kcount * OUT);
  float acc = first ? 0.f : msg[t];
#pragma unroll
  for (int c = 0; c < 8; ++c) {
    int k = kidx8[e * 8 + c] - kbase;
    if ((unsigned)k < (unsigned)kcount)
      acc += basis8[e * 8 + c] * Yn[(size_t)k * OUT + o];
  }
  msg[t] = acc;
}

__global__ void k_msg_scatter(const float* __restrict__ msg,
                              const int* __restrict__ dst,
                              float* __restrict__ agg, int OUT, int E) {
  int t = blockIdx.x * blockDim.x + threadIdx.x;
  if (t >= E * OUT) return;
  int o = t % OUT;
  int e = t / OUT;
  atomicAdd(&agg[dst[e] * OUT + o], msg[t]);
}

// mean-normalize + root matvec + bias + ELU
__global__ void k_finalize(const float* __restrict__ agg,
                           const float* __restrict__ deg,
                           const float* __restrict__ xin,
                           const float* __restrict__ root,
                           const float* __restrict__ bias,
                           float* __restrict__ xout, int N, int IN, int OUT) {
  int t = blockIdx.x * blockDim.x + threadIdx.x;
  if (t >= N * OUT) return;
  int o = t % OUT;
  int n = t / OUT;
  float val = agg[t] / fmaxf(deg[n], 1.f);
  float r = 0.f;
  for (int i = 0; i < IN; ++i) r += xin[n * IN + i] * root[i * OUT + o];
  xout[t] = eluf(val + r + bias[o]);
}

__global__ void k_pool_max(const float* __restrict__ x,
                           const int* __restrict__ cluster,
                           unsigned* __restrict__ pbuf, int N, int OUT) {
  int t = blockIdx.x * blockDim.x + threadIdx.x;
  if (t >= N * OUT) return;
  int o = t % OUT;
  int n = t / OUT;
  atomicMax(&pbuf[cluster[n] * OUT + o], f2ord(x[t]));
}

__global__ void k_pool_dec(const unsigned* __restrict__ pbuf,
                           float* __restrict__ xout, int n) {
  int t = blockIdx.x * blockDim.x + threadIdx.x;
  if (t < n) xout[t] = ord2f(pbuf[t]);
}

// FC1: [16,1024] @ [1024,256] + bias, ELU.  M=16 is one WMMA tile row.
__global__ void k_fc1_wmma(const float* __restrict__ A,
                           const float* __restrict__ Wm,
                           const float* __restrict__ bias,
                           float* __restrict__ out) {
  int nt = blockIdx.x;          // 16 column tiles
  int lane = threadIdx.x;
  int half = lane >> 4;
  int lm = lane & 15;
  int ncol = nt * 16 + lm;
  v8f c = {};
  for (int k0 = 0; k0 < 1024; k0 += 4) {
    int ka = k0 + 2 * half;
    v2f a = { A[(size_t)lm * 1024 + ka], A[(size_t)lm * 1024 + ka + 1] };
    v2f b = { Wm[(size_t)ka * 256 + ncol], Wm[(size_t)(ka + 1) * 256 + ncol] };
    c = __builtin_amdgcn_wmma_f32_16x16x4_f32(false, a, false, b,
                                              (short)0, c, false, false);
  }
#pragma unroll
  for (int v = 0; v < 8; ++v) {
    int m = v + 8 * half;
    out[(size_t)m * 256 + ncol] = eluf(c[v] + bias[ncol]);
  }
}

__global__ void k_fc2(const float* __restrict__ x, const float* __restrict__ Wm,
                      const float* __restrict__ bias, float* __restrict__ out) {
  int t = blockIdx.x * blockDim.x + threadIdx.x;
  if (t >= 16 * 10) return;
  int col = t % 10;
  int row = t / 10;
  float acc = bias[col];
  for (int i = 0; i < 256; ++i) acc += x[row * 256 + i] * Wm[i * 10 + col];
  out[t] = acc;
}

__global__ void k_logsoftmax(const float* __restrict__ x, float* __restrict__ out) {
  int r = blockIdx.x * blockDim.x + threadIdx.x;
  if (r >= 16) return;
  float mx = x[r * 10];
  for (int i = 1; i < 10; ++i) mx = fmaxf(mx, x[r * 10 + i]);
  float s = 0.f;
  for (int i = 0; i < 10; ++i) s += expf(x[r * 10 + i] - mx);
  float lse = mx + logf(s);
  for (int i = 0; i < 10; ++i) out[r * 10 + i] = x[r * 10 + i] - lse;
}

// ---------------------------------------------------------------------------

extern "C" void kernel_launch(void* const* d_in, const int* in_sizes, int n_in,
                              void* d_out, int out_size, void* d_ws,
                              size_t ws_size, hipStream_t stream) {
  const float* x0 = (const float*)d_in[0];
  const float* pseudo[4] = {(const float*)d_in[1], (const float*)d_in[2],
                            (const float*)d_in[3], (const float*)d_in[4]};
  const float* W[4]    = {(const float*)d_in[5], (const float*)d_in[8],
                          (const float*)d_in[11], (const float*)d_in[14]};
  const float* root[4] = {(const float*)d_in[6], (const float*)d_in[9],
                          (const float*)d_in[12], (const float*)d_in[15]};
  const float* bias[4] = {(const float*)d_in[7], (const float*)d_in[10],
                          (const float*)d_in[13], (const float*)d_in[16]};
  const float* fc1_w = (const float*)d_in[17];
  const float* fc1_b = (const float*)d_in[18];
  const float* fc2_w = (const float*)d_in[19];
  const float* fc2_b = (const float*)d_in[20];
  const int* src[4]     = {(const int*)d_in[21], (const int*)d_in[23],
                           (const int*)d_in[25], (const int*)d_in[27]};
  const int* dst[4]     = {(const int*)d_in[22], (const int*)d_in[24],
                           (const int*)d_in[26], (const int*)d_in[28]};
  const int* cluster[4] = {(const int*)d_in[29], (const int*)d_in[30],
                           (const int*)d_in[31], (const int*)d_in[32]};

  // workspace layout (float offsets); total ~65.2 MB
  float* ws = (float*)d_ws;
  float*    XOUT  = ws + 0;          // 524288  (N0*32 max)
  float*    XIN   = ws + 524288;     // 131072  (pooled features)
  float*    AGG   = ws + 655360;     // 524288
  float*    DEG   = ws + 1179648;    // 16384
  unsigned* POOLU = (unsigned*)(ws + 1196032);  // 131072
  int*      KIDX  = (int*)(ws + 1327104);       // 2097152 (E0*8)
  float*    BASIS = ws + 3424256;    // 2097152
  float*    MSG   = ws + 5521408;    // 4194304 (E1*64 max)
  float*    Y     = ws + 9715712;    // 6553600 (N1*25*64 max)
  float*    FC1   = ws + 16269312;   // 4096
  float*    FC2   = ws + 16273408;   // 160

  const int Ns[5]   = {16384, 4096, 1024, 256, 128};
  const int Es[4]   = {262144, 65536, 16384, 4096};
  const int INs[4]  = {1, 32, 64, 64};
  const int OUTs[4] = {32, 64, 64, 128};
  const int KCH = 25;  // 125 spline kernels processed in 5 chunks of 25

  auto cdiv = [](long a, long b) { return (int)((a + b - 1) / b); };

  const float* xin_cur = x0;
  for (int L = 0; L < 4; ++L) {
    int N = Ns[L], E = Es[L], IN = INs[L], OUT = OUTs[L];
    k_zero<<<cdiv((long)N * OUT, 256), 256, 0, stream>>>(AGG, N * OUT);
    k_zero<<<cdiv(N, 256), 256, 0, stream>>>(DEG, N);
    k_edge_prep<<<cdiv(E, 256), 256, 0, stream>>>(pseudo[L], dst[L], E, KIDX,
                                                  BASIS, DEG);
    if (L == 0) {
      k_l1_edge<<<cdiv((long)E * 32, 256), 256, 0, stream>>>(
          x0, src[0], dst[0], W[0], KIDX, BASIS, AGG, E);
    } else {
      for (int ch = 0; ch < 5; ++ch) {
        int kbase = ch * KCH;
        int blocks = (N / 16) * KCH * (OUT / 16);
        k_ygemm<<<blocks, 32, 0, stream>>>(
            xin_cur, W[L] + (size_t)kbase * IN * OUT, Y, IN, OUT, KCH);
        k_msg_accum<<<cdiv((long)E * OUT, 256), 256, 0, stream>>>(
            Y, src[L], KIDX, BASIS, MSG, OUT, kbase, KCH, E, ch == 0 ? 1 : 0);
      }
      k_msg_scatter<<<cdiv((long)E * OUT, 256), 256, 0, stream>>>(MSG, dst[L],
                                                                  AGG, OUT, E);
    }
    k_finalize<<<cdiv((long)N * OUT, 256), 256, 0, stream>>>(
        AGG, DEG, xin_cur, root[L], bias[L], XOUT, N, IN, OUT);
    int Nn = Ns[L + 1];
    k_zero<<<cdiv((long)Nn * OUT, 256), 256, 0, stream>>>((float*)POOLU,
                                                          Nn * OUT);
    k_pool_max<<<cdiv((long)N * OUT, 256), 256, 0, stream>>>(XOUT, cluster[L],
                                                             POOLU, N, OUT);
    k_pool_dec<<<cdiv((long)Nn * OUT, 256), 256, 0, stream>>>(POOLU, XIN,
                                                              Nn * OUT);
    xin_cur = XIN;
  }

  // head: [128,128] pooled == [16,1024] row-major
  k_fc1_wmma<<<16, 32, 0, stream>>>(XIN, fc1_w, fc1_b, FC1);
  k_fc2<<<1, 256, 0, stream>>>(FC1, fc2_w, fc2_b, FC2);
  k_logsoftmax<<<1, 32, 0, stream>>>(FC2, (float*)d_out);
}